// GridCellLoss_42245298323541
// MI455X (gfx1250) — compile-verified
//
#include <hip/hip_runtime.h>
#include <hip/hip_bf16.h>
#include <math.h>

// ---------------------------------------------------------------------------
// GridCellLoss for MI455X (gfx1250, wave32).
// Gram matrix g@g^T in FULL FP32 via v_wmma_f32_16x16x4_f32 (native f32 WMMA,
// bit-faithful to the reference's f32 dot products), transcendental-heavy JSD
// pair term as wave-per-pair VALU work with b128 loads, deterministic
// fixed-order final reduction (no float atomics).
// ---------------------------------------------------------------------------

typedef __attribute__((ext_vector_type(2))) float v2f;
typedef __attribute__((ext_vector_type(4))) float v4f;
typedef __attribute__((ext_vector_type(8))) float v8f;

#define B        4
#define S        128
#define KDIM     512
#define R        32          // logits rows per (b,s)
#define NCOL     32          // logits cols
#define NBS      (B * S)                 // 512
#define NROWS    (NBS * R)               // 16384 softmax rows
#define NPAIRSQ  (B * S * S)             // 65536 (full square, i<j valid)
#define NPAIRS   8128                    // S*(S-1)/2
#define ALPHA_C  0.54f
#define INV2SIG2 (1.0f / (2.0f * 1.2f * 1.2f))

// workspace byte offsets
#define OFF_P       0u                            // 4*128*1024 f32     = 2 MB
#define OFF_AROW    2097152u                      // 4*128*32 f32       = 64 KB
#define OFF_ASUM    2162688u                      // 512 f32
#define OFF_SQ      2164736u                      // 512 f32
#define OFF_ROWSUM  2166784u                      // 512 f32
#define OFF_DIST    2168832u                      // 4*128*128 f32      = 256 KB
#define OFF_TERM    2430976u                      // 4*128*128 f32      = 256 KB
// total ~2.57 MB

__device__ __forceinline__ float wave_sum(float v) {
#pragma unroll
  for (int m = 16; m >= 1; m >>= 1) v += __shfl_xor(v, m, 32);
  return v;
}
__device__ __forceinline__ float wave_max(float v) {
#pragma unroll
  for (int m = 16; m >= 1; m >>= 1) v = fmaxf(v, __shfl_xor(v, m, 32));
  return v;
}

// --- 1) g prep: squared norms + row sums (for dist trick and loss_cap) -----
__global__ void kg_prep(const float* __restrict__ g,
                        float* __restrict__ sq, float* __restrict__ rowsum) {
  int wid  = (blockIdx.x * blockDim.x + threadIdx.x) >> 5;  // (b,s) row
  int lane = threadIdx.x & 31;
  if (wid >= NBS) return;
  const float* row = g + (size_t)wid * KDIM;
  float s2 = 0.0f, s1 = 0.0f;
#pragma unroll
  for (int t0 = 0; t0 < KDIM; t0 += 128) {
    v4f v = *(const v4f*)(row + t0 + lane * 4);
    s2 += v.x * v.x + v.y * v.y + v.z * v.z + v.w * v.w;
    s1 += v.x + v.y + v.z + v.w;
  }
  s2 = wave_sum(s2);
  s1 = wave_sum(s1);
  if (lane == 0) { sq[wid] = s2; rowsum[wid] = s1; }
}

// --- 2) exact log-softmax per 32-wide row; store p and row entropy ---------
__global__ void ksoftmax(const float* __restrict__ logits,
                         float* __restrict__ p, float* __restrict__ arow) {
  int wid  = (blockIdx.x * blockDim.x + threadIdx.x) >> 5;  // (b,s,r) row
  int lane = threadIdx.x & 31;
  if (wid >= NROWS) return;
  float x   = logits[(size_t)wid * NCOL + lane];
  float mx  = wave_max(x);
  float e   = __expf(x - mx);
  float se  = wave_sum(e);
  float lp  = x - mx - __logf(se);
  float pv  = __expf(lp);
  p[(size_t)wid * NCOL + lane] = pv;
  float a = wave_sum(pv * lp);
  if (lane == 0) arow[wid] = a;   // sum_k p*logp for this row
}

// --- 3) Asum[b,s] = sum over 32 rows of row entropies ----------------------
__global__ void kasum(const float* __restrict__ arow, float* __restrict__ asum) {
  int wid  = (blockIdx.x * blockDim.x + threadIdx.x) >> 5;  // (b,s)
  int lane = threadIdx.x & 31;
  if (wid >= NBS) return;
  float a = wave_sum(arow[wid * R + lane]);
  if (lane == 0) asum[wid] = a;
}

// --- 4) Gram via native-f32 WMMA: one wave per 16x16 tile, K=512 ------------
// V_WMMA_F32_16X16X4_F32 fragment layouts (ISA 7.12.2, 32-bit operands):
//   A 16x4 (MxK): lane m=lane&15 is row; lanes 0-15 hold {K=0,K=1},
//                 lanes 16-31 hold {K=2,K=3}  -> v2f = {k+2*half, k+2*half+1}
//   B 4x16 (KxN): lane n=lane&15 is column; same K split by lane half.
//   C/D f32 16x16: vgpr r -> row = r + 8*half, col = lane&15.
__global__ void kgram(const float* __restrict__ g,
                      const float* __restrict__ sq, float* __restrict__ dist) {
  int wid  = (blockIdx.x * blockDim.x + threadIdx.x) >> 5;
  int lane = threadIdx.x & 31;
  if (wid >= B * 64) return;                 // 4 batches x (8x8 tiles)
  int b    = wid >> 6;
  int tile = wid & 63;
  int i0   = (tile >> 3) << 4;
  int j0   = (tile & 7) << 4;
  int m    = lane & 15;
  int half = lane >> 4;

  const float* Abase = g + ((size_t)(b * S + i0 + m) * KDIM) + 2 * half;
  const float* Bbase = g + ((size_t)(b * S + j0 + m) * KDIM) + 2 * half;

  v8f c = {};
#pragma unroll 16
  for (int k = 0; k < KDIM; k += 4) {
    v2f a  = *(const v2f*)(Abase + k);
    v2f bb = *(const v2f*)(Bbase + k);
    c = __builtin_amdgcn_wmma_f32_16x16x4_f32(
        /*neg_a=*/false, a, /*neg_b=*/false, bb,
        /*c_mod=*/(short)0, c, /*reuse_a=*/false, /*reuse_b=*/false);
  }

#pragma unroll
  for (int r = 0; r < 8; r++) {
    int i = i0 + r + 8 * half;
    int j = j0 + m;
    float d2 = sq[b * S + i] + sq[b * S + j] - 2.0f * c[r];
    dist[(size_t)(b * S + i) * S + j] = sqrtf(fmaxf(d2, 0.0f));
  }
}

// --- 5) per-pair JSD + weighted distance term ------------------------------
__global__ void kpair(const float* __restrict__ p, const float* __restrict__ asum,
                      const float* __restrict__ dist, float* __restrict__ term) {
  int wid  = (blockIdx.x * blockDim.x + threadIdx.x) >> 5;
  int lane = threadIdx.x & 31;
  if (wid >= NPAIRSQ) return;
  int j = wid & 127;
  int i = (wid >> 7) & 127;
  int b = wid >> 14;
  size_t idx = (size_t)(b * S + i) * S + j;
  if (i >= j) {                     // invalid half: write zero every call
    if (lane == 0) term[idx] = 0.0f;
    return;
  }
  const float* pi = p + (size_t)(b * S + i) * (R * NCOL);
  const float* pj = p + (size_t)(b * S + j) * (R * NCOL);
  float acc = 0.0f;
#pragma unroll
  for (int t0 = 0; t0 < R * NCOL; t0 += 128) {
    v4f a = *(const v4f*)(pi + t0 + lane * 4);
    v4f c = *(const v4f*)(pj + t0 + lane * 4);
    float s0 = a.x + c.x, s1 = a.y + c.y, s2 = a.z + c.z, s3 = a.w + c.w;
    acc += s0 * __logf(0.5f * s0);
    acc += s1 * __logf(0.5f * s1);
    acc += s2 * __logf(0.5f * s2);
    acc += s3 * __logf(0.5f * s3);
  }
  acc = wave_sum(acc);              // Csum over all 32x32 elements
  if (lane == 0) {
    float jsd = 0.5f * (asum[b * S + i] + asum[b * S + j] - acc) * (1.0f / 32.0f);
    float w   = __expf(-jsd * jsd * INV2SIG2);
    float d   = jsd - dist[idx];
    term[idx] = w * d * d;
  }
}

// --- 6) deterministic fixed-order final reduction --------------------------
__global__ void kfinal(const float* __restrict__ term,
                       const float* __restrict__ rowsum, float* __restrict__ out) {
  __shared__ float sm[256];
  int t = threadIdx.x;
  float s = 0.0f;
  for (int k = t; k < NPAIRSQ; k += 256) s += term[k];
  sm[t] = s;
  __syncthreads();
  for (int off = 128; off > 0; off >>= 1) {
    if (t < off) sm[t] += sm[t + off];
    __syncthreads();
  }
  float sterm = sm[0];
  __syncthreads();
  float sr = 0.0f;
  for (int k = t; k < NBS; k += 256) sr += rowsum[k];
  sm[t] = sr;
  __syncthreads();
  for (int off = 128; off > 0; off >>= 1) {
    if (t < off) sm[t] += sm[t + off];
    __syncthreads();
  }
  if (t == 0) {
    float loss_dist = sterm * (1.0f / (float)(B * NPAIRS));
    float loss_cap  = -sm[0] * (1.0f / (float)NBS);
    out[0] = ALPHA_C * loss_dist + (1.0f - ALPHA_C) * loss_cap;
  }
}

extern "C" void kernel_launch(void* const* d_in, const int* in_sizes, int n_in,
                              void* d_out, int out_size, void* d_ws, size_t ws_size,
                              hipStream_t stream) {
  (void)in_sizes; (void)n_in; (void)out_size; (void)ws_size;
  const float* g      = (const float*)d_in[0];   // (4,128,512) f32
  const float* logits = (const float*)d_in[1];   // (4,128,32,32) f32
  float* out          = (float*)d_out;           // scalar f32

  char* ws = (char*)d_ws;
  float* p       = (float*)(ws + OFF_P);
  float* arow    = (float*)(ws + OFF_AROW);
  float* asum    = (float*)(ws + OFF_ASUM);
  float* sq      = (float*)(ws + OFF_SQ);
  float* rowsum  = (float*)(ws + OFF_ROWSUM);
  float* dist    = (float*)(ws + OFF_DIST);
  float* term    = (float*)(ws + OFF_TERM);

  // 512 waves
  kg_prep<<<dim3(NBS * 32 / 256), dim3(256), 0, stream>>>(g, sq, rowsum);
  // 16384 waves
  ksoftmax<<<dim3(NROWS * 32 / 256), dim3(256), 0, stream>>>(logits, p, arow);
  // 512 waves
  kasum<<<dim3(NBS * 32 / 256), dim3(256), 0, stream>>>(arow, asum);
  // 256 waves (4 batches x 64 tiles), native f32 WMMA
  kgram<<<dim3(B * 64 * 32 / 256), dim3(256), 0, stream>>>(g, sq, dist);
  // 65536 waves
  kpair<<<dim3(NPAIRSQ * 32 / 256), dim3(256), 0, stream>>>(p, asum, dist, term);
  // single block, fixed-order -> deterministic
  kfinal<<<dim3(1), dim3(256), 0, stream>>>(term, rowsum, out);
}